// PeakSuppress_67834713473747
// MI455X (gfx1250) — compile-verified
//
#include <hip/hip_runtime.h>

typedef __attribute__((ext_vector_type(2))) float v2f;
typedef __attribute__((ext_vector_type(4))) float v4f;
typedef __attribute__((ext_vector_type(8))) float v8f;

#define NB    32
#define NC    256
#define HW    4096
#define KDROP 1024

// ---------------------------------------------------------------------------
// Kernel 1: channel-sum reduction expressed as WMMA.
//   sums[b][p] = sum_c features[b][c][p]
// A = ones(16x4) fp32, B = 4 channels x 16 pixels tile, D accumulated over
// 64 steps (K=4 each) -> exact fp32 sums via v_wmma_f32_16x16x4_f32.
// Each wave (32 lanes) produces 16 pixel sums. EXEC is all-1s through the
// WMMA loop (full wave, no divergence); the predicated store comes after.
// ---------------------------------------------------------------------------
__global__ __launch_bounds__(256) void PeakSuppress_sum_wmma(
    const float* __restrict__ f, float* __restrict__ sums) {
  const int lane = threadIdx.x & 31;
  const int wave = threadIdx.x >> 5;
  const int b    = blockIdx.x >> 5;        // 32 blocks per batch image
  const int pg   = blockIdx.x & 31;        // 32 pixel-groups of 128
  const int pbase = pg * 128 + wave * 16;  // this wave's 16 pixels
  const int n  = lane & 15;                // column (pixel) in the 16-wide tile
  const int kb = (lane >> 4) << 1;         // channel sub-slot 0 or 2

  const float* base = f + ((size_t)b * NC + kb) * HW + pbase + n;

  v2f a; a.x = 1.0f; a.y = 1.0f;           // ones -> D cols = channel sums
  v8f acc = {};
#pragma unroll 8
  for (int c0 = 0; c0 < NC; c0 += 4) {
    v2f bv;
    bv.x = base[0];
    bv.y = base[HW];
    base += 4 * (size_t)HW;
    acc = __builtin_amdgcn_wmma_f32_16x16x4_f32(
        /*neg_a=*/false, a, /*neg_b=*/false, bv,
        /*c_mod=*/(short)0, acc, /*reuse_a=*/false, /*reuse_b=*/false);
  }
  // D VGPR0, lanes 0-15 = (M=0, N=lane): the 16 pixel sums.
  if (lane < 16) sums[(size_t)b * HW + pbase + lane] = acc[0];
}

// ---------------------------------------------------------------------------
// Kernel 2: per-row exact 1024th-largest via 4-pass radix select in LDS,
// then write the 0/1 mask in place (block owns its whole row).
// Deterministic: histogram adds commute; ties dropped smallest-index-first
// (matches jax.lax.top_k stable ordering).
// ---------------------------------------------------------------------------
__global__ __launch_bounds__(256) void PeakSuppress_select(
    const float* __restrict__ sums, float* __restrict__ mask) {
  __shared__ unsigned keys[HW];
  __shared__ unsigned hist[256];
  __shared__ unsigned scanbuf[256];
  __shared__ unsigned s_prefix, s_remaining, s_greater;

  const int t   = threadIdx.x;
  const int row = blockIdx.x;
  const float* srow = sums + (size_t)row * HW;

  // Order-preserving float -> u32 key (larger float => larger key).
  for (int i = t; i < HW; i += 256) {
    unsigned bits = __float_as_uint(srow[i]);
    keys[i] = (bits & 0x80000000u) ? ~bits : (bits | 0x80000000u);
  }
  if (t == 0) { s_prefix = 0u; s_remaining = KDROP; s_greater = 0u; }
  __syncthreads();

  // MSB-first radix select for the KDROP-th largest key.
  for (int pos = 3; pos >= 0; --pos) {
    hist[t] = 0u;
    __syncthreads();
    const unsigned prefix = s_prefix;
    const int shiftHi = 8 * pos + 8;
    for (int i = t; i < HW; i += 256) {
      const unsigned key = keys[i];
      const bool active = (pos == 3) || (((key ^ prefix) >> shiftHi) == 0u);
      if (active) atomicAdd(&hist[(key >> (8 * pos)) & 255u], 1u);
    }
    __syncthreads();
    if (t == 0) {
      unsigned rem = s_remaining;
      int bin = 255;
      for (; bin > 0; --bin) {
        const unsigned h = hist[bin];
        if (rem <= h) break;
        rem -= h;
      }
      s_prefix    = prefix | ((unsigned)bin << (8 * pos));
      s_remaining = rem;
    }
    __syncthreads();
  }
  const unsigned thr = s_prefix;

  // Count strictly-greater, and rank ties by index for an exact quota.
  const int CHUNK = HW / 256;  // 16 contiguous elements per thread
  const int basei = t * CHUNK;
  unsigned localGt = 0, localEq = 0;
  for (int i = 0; i < CHUNK; ++i) {
    const unsigned key = keys[basei + i];
    localGt += (key > thr) ? 1u : 0u;
    localEq += (key == thr) ? 1u : 0u;
  }
  atomicAdd(&s_greater, localGt);
  scanbuf[t] = localEq;
  __syncthreads();
  // Hillis-Steele inclusive scan over the 256 per-thread tie counts.
  for (int off = 1; off < 256; off <<= 1) {
    const unsigned mine = scanbuf[t];
    const unsigned add  = (t >= off) ? scanbuf[t - off] : 0u;
    __syncthreads();
    scanbuf[t] = mine + add;
    __syncthreads();
  }
  const unsigned eqBefore = scanbuf[t] - localEq;  // exclusive prefix
  __syncthreads();
  const unsigned quota = KDROP - s_greater;  // ties to drop (index order)

  float* mrow = mask + (size_t)row * HW;     // in-place over sums: safe,
  unsigned eqSeen = eqBefore;                // all reads done (keys in LDS)
  for (int i = 0; i < CHUNK; ++i) {
    const unsigned key = keys[basei + i];
    float m = 1.0f;
    if (key > thr) m = 0.0f;
    else if (key == thr) { if (eqSeen < quota) m = 0.0f; ++eqSeen; }
    mrow[basei + i] = m;
  }
}

// ---------------------------------------------------------------------------
// Kernel 3: out = features * mask[b][p], float4 (global_load/store_b128).
// Second read of features should largely hit the 192MB L2.
// ---------------------------------------------------------------------------
__global__ __launch_bounds__(256) void PeakSuppress_apply(
    const float* __restrict__ f, const float* __restrict__ mask,
    float* __restrict__ out) {
  const size_t idx = (size_t)blockIdx.x * 256 + threadIdx.x;
  const size_t e0  = idx * 4;                 // flat element index
  const size_t b   = e0 >> 20;                // NC*HW = 1<<20
  const size_t p   = e0 & (HW - 1);           // pixel within row (mult of 4)
  const v4f v = ((const v4f*)f)[idx];
  const v4f m = *(const v4f*)(mask + b * HW + p);
  ((v4f*)out)[idx] = v * m;
}

extern "C" void kernel_launch(void* const* d_in, const int* in_sizes, int n_in,
                              void* d_out, int out_size, void* d_ws, size_t ws_size,
                              hipStream_t stream) {
  const float* f = (const float*)d_in[0];
  float* out = (float*)d_out;
  float* ws  = (float*)d_ws;  // 32*4096 floats = 512 KiB: sums, then mask in-place

  // 1) channel sums via WMMA: 1024 blocks x 256 (8 waves of 16 pixels each)
  PeakSuppress_sum_wmma<<<NB * (HW / 128), 256, 0, stream>>>(f, ws);
  // 2) exact top-1024 threshold + mask, one block per row
  PeakSuppress_select<<<NB, 256, 0, stream>>>(ws, ws);
  // 3) masked copy, float4: 33.5M elems / 4 / 256 = 32768 blocks
  PeakSuppress_apply<<<(NB * NC * HW) / 4 / 256, 256, 0, stream>>>(f, ws, out);
}